// GraphSage_85676007621255
// MI455X (gfx1250) — compile-verified
//
#include <hip/hip_runtime.h>

#define EPSV       1e-5f
#define NEG_SLOPE  0.01f

#define N0_  100000
#define N1_  25000
#define N2_  5000
#define E0_  800000
#define E1_  160000
#define D_IN  256
#define D_HID 256
#define D_OUT 128

typedef __attribute__((ext_vector_type(2))) float v2f;
typedef __attribute__((ext_vector_type(8))) float v8f;

// ---------------------------------------------------------------------------
// Edge scatter: one wave (32 lanes) per edge. Gathers X[src[e]] (D floats),
// atomically accumulates into agg[dst[e]], bumps deg[dst[e]]. Optionally
// writes the time-mask (as 0.0/1.0 float) to maskOut[e].
// ---------------------------------------------------------------------------
__global__ void scatter_edges(const float* __restrict__ X,
                              const int* __restrict__ src,
                              const int* __restrict__ dst,
                              const int* __restrict__ val,
                              const int* __restrict__ ts,
                              const int* __restrict__ timeP,
                              const int* __restrict__ intervalP,
                              float* __restrict__ agg,
                              float* __restrict__ deg,
                              int E, int D,
                              float* __restrict__ maskOut)
{
    int e = blockIdx.x * blockDim.y + threadIdx.y;
    if (e >= E) return;
    int lane = threadIdx.x;

    int t0 = *timeP;
    int iv = *intervalP;
    int t  = ts[val[e]];
    bool m = (t >= t0) && (t < t0 + iv);
    if (maskOut != nullptr && lane == 0) maskOut[e] = m ? 1.0f : 0.0f;
    if (!m) return;

    int s = src[e];
    int d = dst[e];
    const float* xr = X + (size_t)s * D;
    float*       ar = agg + (size_t)d * D;
    for (int j = lane; j < D; j += 32)
        atomicAdd(&ar[j], xr[j]);
    if (lane == 0) atomicAdd(&deg[d], 1.0f);
}

// rdeg[i] = 1 / max(deg[i], 1)
__global__ void rdeg_kernel(const float* __restrict__ deg,
                            float* __restrict__ rdeg, int n)
{
    int i = blockIdx.x * blockDim.x + threadIdx.x;
    if (i < n) rdeg[i] = 1.0f / fmaxf(deg[i], 1.0f);
}

// ---------------------------------------------------------------------------
// Fused SAGE GEMM, V_WMMA_F32_16X16X4_F32, one wave = one 16x16 C tile.
//   C = (Aagg * rdeg[row]) @ Wl + Aroot @ Wr + bias [+ BN(eval)+leaky_relu]
//
// K, N are compile-time so all B loads are immediate-offset global_load_b32
// off a single base, and A loads are immediate-offset global_load_b64.
// Out-of-range rows are handled branchlessly: row index clamped to a valid
// address, contribution zeroed via a per-lane scale (keeps EXEC uniform for
// WMMA and avoids saveexec divergence in the hot loop).
// K is processed in chunks of 16 (4 WMMA steps): 12 clause-able loads, then
// 4 back-to-back WMMAs, so loads of chunk i+1 overlap WMMAs of chunk i.
// ---------------------------------------------------------------------------
template <int K, int N, int DOBN>
__global__ void __launch_bounds__(128)
sage_gemm_wmma(const float* __restrict__ Aagg,
               const float* __restrict__ rdeg,
               const float* __restrict__ Aroot,
               const float* __restrict__ Wl,
               const float* __restrict__ Wr,
               const float* __restrict__ bias,
               const float* __restrict__ g,
               const float* __restrict__ bt,
               const float* __restrict__ rm,
               const float* __restrict__ rv,
               float* __restrict__ Cout,
               int M)
{
    const int lane = threadIdx.x;          // 0..31
    const int half = lane >> 4;            // 0: K=k..k+1, 1: K=k+2..k+3
    const int l15  = lane & 15;
    const int col  = (blockIdx.x * blockDim.y + threadIdx.y) * 16 + l15;
    const int rowA = blockIdx.y * 16 + l15;

    const bool rowOK  = (rowA < M);
    const int  rowClp = rowOK ? rowA : (M - 1);       // safe address
    const float scAgg  = rowOK ? rdeg[rowClp] : 0.0f; // mean-norm, 0 if OOB
    const float scRoot = rowOK ? 1.0f : 0.0f;

    // bases include the lane-half K offset; all further offsets are constants
    const float* a1 = Aagg  + (size_t)rowClp * K + half * 2;
    const float* a2 = Aroot + (size_t)rowClp * K + half * 2;
    const float* bl = Wl + (size_t)(half * 2) * N + col;
    const float* br = Wr + (size_t)(half * 2) * N + col;

    v8f c = {};

#pragma unroll
    for (int k0 = 0; k0 < K; k0 += 16) {
        v2f a[4], b[4];
#pragma unroll
        for (int u = 0; u < 4; ++u) {
            const int kk = k0 + u * 4;
            a[u]   = *(const v2f*)(a1 + kk);          // global_load_b64, imm off
            b[u].x = bl[(size_t)kk * N];              // imm off
            b[u].y = bl[(size_t)(kk + 1) * N];        // imm off
        }
#pragma unroll
        for (int u = 0; u < 4; ++u) {
            v2f aa; aa.x = a[u].x * scAgg; aa.y = a[u].y * scAgg;
            c = __builtin_amdgcn_wmma_f32_16x16x4_f32(false, aa, false, b[u],
                                                      (short)0, c, false, false);
        }
    }
#pragma unroll
    for (int k0 = 0; k0 < K; k0 += 16) {
        v2f a[4], b[4];
#pragma unroll
        for (int u = 0; u < 4; ++u) {
            const int kk = k0 + u * 4;
            a[u]   = *(const v2f*)(a2 + kk);
            b[u].x = br[(size_t)kk * N];
            b[u].y = br[(size_t)(kk + 1) * N];
        }
#pragma unroll
        for (int u = 0; u < 4; ++u) {
            v2f aa; aa.x = a[u].x * scRoot; aa.y = a[u].y * scRoot;
            c = __builtin_amdgcn_wmma_f32_16x16x4_f32(false, aa, false, b[u],
                                                      (short)0, c, false, false);
        }
    }

    // epilogue: bias (+ BatchNorm eval + leaky_relu for layer 1)
    const float bi = bias[col];
    float gg = 1.0f, bb = 0.0f, mmn = 0.0f, ivr = 1.0f;
    if (DOBN) {
        gg  = g[col];
        bb  = bt[col];
        mmn = rm[col];
        ivr = rsqrtf(rv[col] + EPSV);
    }
    const int rowBase = blockIdx.y * 16 + half * 8;
    float* outBase = Cout + (size_t)rowBase * N + col;
#pragma unroll
    for (int i = 0; i < 8; ++i) {
        if (rowBase + i < M) {
            float v = c[i] + bi;
            if (DOBN) {
                v = (v - mmn) * ivr * gg + bb;
                v = (v >= 0.0f) ? v : NEG_SLOPE * v;
            }
            outBase[(size_t)i * N] = v;   // imm-offset stores
        }
    }
}

// ---------------------------------------------------------------------------
extern "C" void kernel_launch(void* const* d_in, const int* in_sizes, int n_in,
                              void* d_out, int out_size, void* d_ws, size_t ws_size,
                              hipStream_t stream)
{
    (void)in_sizes; (void)n_in; (void)out_size; (void)ws_size;

    const float* x    = (const float*)d_in[0];
    const int*   src0 = (const int*)  d_in[1];
    const int*   dst0 = (const int*)  d_in[2];
    const int*   val0 = (const int*)  d_in[3];
    const int*   src1 = (const int*)  d_in[4];
    const int*   dst1 = (const int*)  d_in[5];
    const int*   val1 = (const int*)  d_in[6];
    const int*   ts   = (const int*)  d_in[7];
    const int*   timeP= (const int*)  d_in[8];
    const int*   intP = (const int*)  d_in[9];
    const float* W1l  = (const float*)d_in[10];
    const float* W1r  = (const float*)d_in[11];
    const float* b1   = (const float*)d_in[12];
    const float* g1   = (const float*)d_in[13];
    const float* bt1  = (const float*)d_in[14];
    const float* rm1  = (const float*)d_in[15];
    const float* rv1  = (const float*)d_in[16];
    const float* W2l  = (const float*)d_in[17];
    const float* W2r  = (const float*)d_in[18];
    const float* b2   = (const float*)d_in[19];

    // workspace layout (floats)
    float* ws = (float*)d_ws;
    size_t off = 0;
    float* agg0  = ws + off; off += (size_t)N1_ * D_IN;   // neighbor sums L0
    float* deg0  = ws + off; off += N1_;                  // degrees L0
    float* rdeg0 = ws + off; off += N1_;
    float* h     = ws + off; off += (size_t)N1_ * D_HID;  // hidden features
    float* agg1  = ws + off; off += (size_t)N2_ * D_HID;  // neighbor sums L1
    float* deg1  = ws + off; off += N2_;
    float* rdeg1 = ws + off; off += N2_;

    // zero the accumulators (agg+deg are contiguous per layer)
    hipMemsetAsync(agg0, 0, ((size_t)N1_ * D_IN  + N1_) * sizeof(float), stream);
    hipMemsetAsync(agg1, 0, ((size_t)N2_ * D_HID + N2_) * sizeof(float), stream);

    dim3 sblk(32, 8);   // 8 edges per 256-thread block
    dim3 gblk(32, 4);   // 4 WMMA waves per block

    // ---- layer 0: scatter x -> agg0/deg0 over E0 edges ----
    scatter_edges<<<dim3((E0_ + 7) / 8), sblk, 0, stream>>>(
        x, src0, dst0, val0, ts, timeP, intP, agg0, deg0, E0_, D_IN, nullptr);
    rdeg_kernel<<<(N1_ + 255) / 256, 256, 0, stream>>>(deg0, rdeg0, N1_);

    // h = mean(agg0) @ W1l + x[:N1] @ W1r + b1, BN + leaky_relu
    sage_gemm_wmma<D_IN, D_HID, 1>
        <<<dim3(D_HID / 64, (N1_ + 15) / 16), gblk, 0, stream>>>(
            agg0, rdeg0, x, W1l, W1r, b1, g1, bt1, rm1, rv1, h, N1_);

    // ---- layer 1: scatter h -> agg1/deg1 over E1 edges; emit mask1 ----
    float* maskOut = (float*)d_out + (size_t)N2_ * D_OUT;
    scatter_edges<<<dim3((E1_ + 7) / 8), sblk, 0, stream>>>(
        h, src1, dst1, val1, ts, timeP, intP, agg1, deg1, E1_, D_HID, maskOut);
    rdeg_kernel<<<(N2_ + 255) / 256, 256, 0, stream>>>(deg1, rdeg1, N2_);

    // out = mean(agg1) @ W2l + h[:N2] @ W2r + b2
    sage_gemm_wmma<D_HID, D_OUT, 0>
        <<<dim3(D_OUT / 64, (N2_ + 15) / 16), gblk, 0, stream>>>(
            agg1, rdeg1, h, W2l, W2r, b2, nullptr, nullptr, nullptr, nullptr,
            (float*)d_out, N2_);
}